// GNNModel_33603824124188
// MI455X (gfx1250) — compile-verified
//
#include <hip/hip_runtime.h>
#include <hip/hip_bf16.h>

// GNN (3x GAT + MLP) for MI455X / gfx1250, wave32.
// GEMMs: v_wmma_f32_16x16x4_f32 with async LDS staging (when available).
// Edge phase: native fp32/u32 global atomics, scalarized wave-uniform loads.

typedef __attribute__((ext_vector_type(2))) float v2f;
typedef __attribute__((ext_vector_type(8))) float v8f;
typedef int v4i __attribute__((vector_size(16)));
typedef __attribute__((address_space(1))) v4i* gv4i_p;  // global (AS1) 16B vector ptr
typedef __attribute__((address_space(3))) v4i* lv4i_p;  // LDS (AS3) 16B vector ptr

#define HAS_ASYNC_LDS __has_builtin(__builtin_amdgcn_global_load_async_to_lds_b128)

__device__ __forceinline__ void wait_asynccnt0() {
#if __has_builtin(__builtin_amdgcn_s_wait_asynccnt)
  __builtin_amdgcn_s_wait_asynccnt(0);
#else
  asm volatile("s_wait_asynccnt 0x0" ::: "memory");
#endif
}

// ---- monotonic float <-> uint order mapping (for atomic segment max) ----
__device__ __forceinline__ unsigned f2ord(float f) {
  unsigned b = __float_as_uint(f);
  return (b & 0x80000000u) ? ~b : (b | 0x80000000u);
}
__device__ __forceinline__ float ord2f(unsigned u) {
  unsigned b = (u & 0x80000000u) ? (u & 0x7FFFFFFFu) : ~u;
  return __uint_as_float(b);
}

// ---- WMMA GEMM: Y[n_rows, NCOLS] = act(X[n_rows,64] @ W[64,NCOLS] + bias) ----
// Block = 256 threads = 8 waves; each wave computes a 16-row stripe.
// A frag (16x4 f32): lane l (m=l&15, half=l>>4): vgpr j = X[m][k0 + half*2 + j]
// B frag (4x16 f32): lane l (n=l&15, half=l>>4): vgpr j = W[k0 + half*2 + j][n]
// C/D (16x16 f32):   lane l (n=l&15): vgpr j = D[(l>=16?8:0)+j][n]
template <int NCOLS, bool RELU, bool HASBIAS>
__global__ __launch_bounds__(256) void gemm64_wmma(
    const float* __restrict__ X, const float* __restrict__ W,
    const float* __restrict__ bias, float* __restrict__ Y, int n_rows) {
  constexpr int NT = NCOLS / 16;
  constexpr int XPITCH = 68;  // 68 floats = 272 B = 17*16 B: conflict-free & 16B-aligned rows
  __shared__ float sW[64 * NCOLS];
  __shared__ float sX[8][16 * XPITCH];

  const int tid = threadIdx.x;
  const int wv = tid >> 5;
  const int lane = tid & 31;
  const int row0 = blockIdx.x * 128 + wv * 16;
  float* sx = sX[wv];

#if HAS_ASYNC_LDS
  // Stage this wave's 16x64 X tile with async global->LDS B128 copies
  // (8 chunks of 16B per lane; ASYNCcnt-tracked, per-wave wait below).
#pragma unroll
  for (int i = 0; i < 8; ++i) {
    int c = lane + i * 32;     // 0..255 : 16B chunk index within tile
    int rrow = c >> 4;         // 0..15
    int col = (c & 15) << 2;   // 0,4,...,60
    int r = row0 + rrow;
    if (r >= n_rows) r = n_rows - 1;  // clamp (stores guarded later)
    float* gp = const_cast<float*>(X + (size_t)r * 64 + col);
    float* lp = sx + rrow * XPITCH + col;
    __builtin_amdgcn_global_load_async_to_lds_b128((gv4i_p)gp, (lv4i_p)lp, 0, 0);
  }
#else
  for (int i = lane; i < 16 * 64; i += 32) {
    int r = row0 + (i >> 6);
    if (r >= n_rows) r = n_rows - 1;
    sx[(i >> 6) * XPITCH + (i & 63)] = X[(size_t)r * 64 + (i & 63)];
  }
#endif

  for (int i = tid; i < 64 * NCOLS; i += 256) sW[i] = W[i];
  __syncthreads();  // sW visibility across waves
#if HAS_ASYNC_LDS
  wait_asynccnt0();  // this wave's sX tile landed
#endif

  const int half = lane >> 4;
  const int nl = lane & 15;
  const int kbase = half * 2;
  const int mbase = half * 8;

  v8f acc[NT];
#pragma unroll
  for (int t = 0; t < NT; ++t) acc[t] = {};

#pragma unroll
  for (int k0 = 0; k0 < 64; k0 += 4) {
    v2f a;
    a.x = sx[nl * XPITCH + k0 + kbase + 0];
    a.y = sx[nl * XPITCH + k0 + kbase + 1];
#pragma unroll
    for (int t = 0; t < NT; ++t) {
      v2f b;
      b.x = sW[(k0 + kbase + 0) * NCOLS + t * 16 + nl];
      b.y = sW[(k0 + kbase + 1) * NCOLS + t * 16 + nl];
      acc[t] = __builtin_amdgcn_wmma_f32_16x16x4_f32(
          false, a, false, b, (short)0, acc[t], false, false);
    }
  }

  // Epilogue: wave-uniform fast path for full tiles.
  const bool full = (row0 + 16) <= n_rows;
#pragma unroll
  for (int t = 0; t < NT; ++t) {
    const int n = t * 16 + nl;
    float bv = 0.0f;
    if (HASBIAS) bv = bias[n];
#pragma unroll
    for (int j = 0; j < 8; ++j) {
      int r = row0 + mbase + j;
      float v = acc[t][j];
      if (HASBIAS) v += bv;
      if (RELU) v = v > 0.0f ? v : 0.0f;
      if (full) {
        Y[(size_t)r * NCOLS + n] = v;
      } else if (r < n_rows) {
        Y[(size_t)r * NCOLS + n] = v;
      }
    }
  }
}

// ---- per-node attention logits: als[i]=dot(h[i],a_src), ald[i]=dot(h[i],a_dst) ----
__global__ __launch_bounds__(256) void node_alpha(
    const float* __restrict__ h, const float* __restrict__ a_src,
    const float* __restrict__ a_dst, float* __restrict__ als,
    float* __restrict__ ald, int N) {
  int wave = (blockIdx.x * 256 + threadIdx.x) >> 5;
  int lane = threadIdx.x & 31;
  if (wave >= N) return;
  float h0 = h[(size_t)wave * 64 + lane];
  float h1 = h[(size_t)wave * 64 + lane + 32];
  float s = h0 * a_src[lane] + h1 * a_src[lane + 32];
  float d = h0 * a_dst[lane] + h1 * a_dst[lane + 32];
#pragma unroll
  for (int off = 16; off > 0; off >>= 1) {
    s += __shfl_xor(s, off, 32);
    d += __shfl_xor(d, off, 32);
  }
  if (lane == 0) {
    als[wave] = s;
    ald[wave] = d;
  }
}

// ---- zero per-layer accumulators (d_ws is poisoned, so explicit init) ----
__global__ __launch_bounds__(256) void init_layer(
    float4* __restrict__ acc4, float* __restrict__ denom,
    unsigned* __restrict__ maxu, int N) {
  int i = blockIdx.x * 256 + threadIdx.x;
  if (i < N * 16) acc4[i] = make_float4(0.f, 0.f, 0.f, 0.f);
  if (i < N) {
    denom[i] = 0.0f;
    maxu[i] = 0u;  // below every encodable finite/inf value
  }
}

// ---- pass 1: e = leaky_relu(as[src]+ad[dst]); segment max over dst ----
__global__ __launch_bounds__(256) void edge_pass1(
    const int* __restrict__ src, const int* __restrict__ dst,
    const float* __restrict__ als, const float* __restrict__ ald,
    float* __restrict__ ebuf, unsigned* __restrict__ maxu, int E) {
  int i = blockIdx.x * 256 + threadIdx.x;
  if (i >= E) return;
  int s = src[i], d = dst[i];
  float v = als[s] + ald[d];
  v = v > 0.0f ? v : 0.2f * v;  // leaky_relu, slope 0.2
  ebuf[i] = v;
  atomicMax(&maxu[d], f2ord(v));
}

// ---- pass 2: w = exp(e - m[dst]); segment sum over dst ----
__global__ __launch_bounds__(256) void edge_pass2(
    const int* __restrict__ dst, float* __restrict__ ebuf,
    const unsigned* __restrict__ maxu, float* __restrict__ denom, int E) {
  int i = blockIdx.x * 256 + threadIdx.x;
  if (i >= E) return;
  int d = dst[i];
  float w = __expf(ebuf[i] - ord2f(maxu[d]));
  ebuf[i] = w;
  atomicAdd(&denom[d], w);
}

// ---- pass 3: acc[dst] += (w/denom[dst]) * h[src]  (one wave per edge) ----
// Edge id forced wave-uniform via readfirstlane so src/dst/w/denom become
// scalar (SMEM) loads instead of 32 redundant vector loads per wave.
__global__ __launch_bounds__(256) void edge_aggregate(
    const int* __restrict__ src, const int* __restrict__ dst,
    const float* __restrict__ wbuf, const float* __restrict__ denom,
    const float* __restrict__ h, float* __restrict__ acc, int E) {
  const int lane = threadIdx.x & 31;
  const int e = __builtin_amdgcn_readfirstlane(blockIdx.x * 8 + (threadIdx.x >> 5));
  if (e >= E) return;
  const int s = src[e];
  const int d = dst[e];
  const float alpha = wbuf[e] / denom[d];
  float h0 = h[(size_t)s * 64 + lane];
  float h1 = h[(size_t)s * 64 + lane + 32];
  atomicAdd(&acc[(size_t)d * 64 + lane], alpha * h0);
  atomicAdd(&acc[(size_t)d * 64 + lane + 32], alpha * h1);
}

// ---- x_next = relu(acc + b), vectorized float4 ----
__global__ __launch_bounds__(256) void finalize_relu(
    const float4* __restrict__ acc4, const float4* __restrict__ bias4,
    float4* __restrict__ xout4, int N) {
  int i = blockIdx.x * 256 + threadIdx.x;
  if (i >= N * 16) return;
  float4 a = acc4[i];
  float4 b = bias4[i & 15];  // 16 float4 per 64-wide row
  float4 r;
  r.x = a.x + b.x; r.x = r.x > 0.f ? r.x : 0.f;
  r.y = a.y + b.y; r.y = r.y > 0.f ? r.y : 0.f;
  r.z = a.z + b.z; r.z = r.z > 0.f ? r.z : 0.f;
  r.w = a.w + b.w; r.w = r.w > 0.f ? r.w : 0.f;
  xout4[i] = r;
}

extern "C" void kernel_launch(void* const* d_in, const int* in_sizes, int n_in,
                              void* d_out, int out_size, void* d_ws, size_t ws_size,
                              hipStream_t stream) {
  const int N = in_sizes[0] / 64;
  const int E = in_sizes[1] / 2;

  const float* x = (const float*)d_in[0];
  const int* ei = (const int*)d_in[1];
  const int* src = ei;
  const int* dst = ei + E;
  const float* W[3] = {(const float*)d_in[2], (const float*)d_in[6], (const float*)d_in[10]};
  const float* As[3] = {(const float*)d_in[3], (const float*)d_in[7], (const float*)d_in[11]};
  const float* Ad[3] = {(const float*)d_in[4], (const float*)d_in[8], (const float*)d_in[12]};
  const float* B[3] = {(const float*)d_in[5], (const float*)d_in[9], (const float*)d_in[13]};
  const float* lin1W = (const float*)d_in[14];
  const float* lin1b = (const float*)d_in[15];
  const float* lin2W = (const float*)d_in[16];
  const float* lin2b = (const float*)d_in[17];

  // workspace (floats): P(N*64) | Q(N*64) | ACC(N*64) | ALS(N) | ALD(N) | DEN(N) | MAXU(N) | EBUF(E)
  float* ws = (float*)d_ws;
  float* P = ws;
  float* Q = P + (size_t)N * 64;
  float* ACC = Q + (size_t)N * 64;
  float* ALS = ACC + (size_t)N * 64;
  float* ALD = ALS + N;
  float* DEN = ALD + N;
  unsigned* MAXU = (unsigned*)(DEN + N);
  float* EBUF = (float*)(MAXU + N);

  dim3 blk(256);
  const int gemm_grid = (N + 127) / 128;
  const int vec4_grid = (N * 16 + 255) / 256;
  const int node_wave_grid = (N * 32 + 255) / 256;
  const int edge_grid = (E + 255) / 256;
  const int agg_grid = (E + 7) / 8;  // one wave (32 lanes) per edge, 8 waves/block

  const float* xin = x;
  for (int l = 0; l < 3; ++l) {
    gemm64_wmma<64, false, false><<<gemm_grid, blk, 0, stream>>>(xin, W[l], nullptr, P, N);
    node_alpha<<<node_wave_grid, blk, 0, stream>>>(P, As[l], Ad[l], ALS, ALD, N);
    init_layer<<<vec4_grid, blk, 0, stream>>>((float4*)ACC, DEN, MAXU, N);
    edge_pass1<<<edge_grid, blk, 0, stream>>>(src, dst, ALS, ALD, EBUF, MAXU, E);
    edge_pass2<<<edge_grid, blk, 0, stream>>>(dst, EBUF, MAXU, DEN, E);
    edge_aggregate<<<agg_grid, blk, 0, stream>>>(src, dst, EBUF, DEN, P, ACC, E);
    finalize_relu<<<vec4_grid, blk, 0, stream>>>((const float4*)ACC, (const float4*)B[l],
                                                 (float4*)Q, N);
    xin = Q;
  }

  // MLP head: relu(Q @ lin1W + lin1b) -> P ; P @ lin2W + lin2b -> d_out
  gemm64_wmma<64, true, true><<<gemm_grid, blk, 0, stream>>>(Q, lin1W, lin1b, P, N);
  gemm64_wmma<32, false, true><<<gemm_grid, blk, 0, stream>>>(P, lin2W, lin2b, (float*)d_out, N);
}